// MLFFNet_4174708212055
// MI455X (gfx1250) — compile-verified
//
#include <hip/hip_runtime.h>

typedef __attribute__((ext_vector_type(2))) float v2f;
typedef __attribute__((ext_vector_type(8))) float v8f;

#define BATCH 32
#define NATOM 216
#define NAT_T 108
#define KNEI  100
#define FDIM  42
#define H0DIM 60
#define H1DIM 30
#define ROWS_T (BATCH * NAT_T)   // 3456 rows per atom type
#define TILES_T (ROWS_T / 16)    // 216 tiles of 16 rows

// ---------------------------------------------------------------------------
// Kernel 1: full MLP forward + backward for one 16-atom tile, one wave32/block.
// All operands are staged zero-padded in LDS so every WMMA fragment load is an
// unconditional, 8B-aligned ds_load_b64 (no EXEC-masked scalar loads).
// fp32 WMMA V_WMMA_F32_16X16X4_F32; A: lanes 0-15 = M, K pair {0,1}; lanes
// 16-31 = K pair {2,3}. C/D: VGPR r = row r / r+8, col = lane&15.
// ---------------------------------------------------------------------------
__global__ __launch_bounds__(32)
void mlp_fwd_bwd(const float* __restrict__ image,
                 const float* __restrict__ w0, const float* __restrict__ b0,
                 const float* __restrict__ w1, const float* __restrict__ b1,
                 const float* __restrict__ wo, const float* __restrict__ bo,
                 float* __restrict__ g,       // [B,N,F] workspace
                 float* __restrict__ ei_out,  // [B,N] section of d_out
                 int toff)
{
    // padded weight tiles, both orientations so B fragments are contiguous
    __shared__ __align__(16) float sW0c[64][64];  // sW0c[n][k] = W0[k][n]  (GEMM1)
    __shared__ __align__(16) float sW0r[48][64];  // sW0r[f][h] = W0[f][h]  (GEMM4: W0^T)
    __shared__ __align__(16) float sW1c[32][64];  // sW1c[n][k] = W1[k][n]  (GEMM2)
    __shared__ __align__(16) float sW1r[64][32];  // sW1r[p][q] = W1[p][q]  (GEMM3: W1^T)
    __shared__ __align__(16) float sX  [16][44];  // X tile, F padded 42->44
    __shared__ __align__(16) float sH0 [16][64];
    __shared__ __align__(16) float sH1 [16][32];
    __shared__ __align__(16) float sdH1[16][32];
    __shared__ __align__(16) float sdH0[16][64];
    __shared__ __align__(16) float sG  [16][48];  // g tile before store
    __shared__ float sb0[64], sb1[32], swo[32];

    const int lane = threadIdx.x;       // 0..31
    const int hf   = lane >> 4;         // K-pair select
    const int lid  = lane & 15;
    const int tile = blockIdx.x;        // 0..215

    // ---- stage: zero padded regions ----
    for (int i = lane; i < 64 * 64; i += 32) (&sW0c[0][0])[i] = 0.f;
    for (int i = lane; i < 48 * 64; i += 32) (&sW0r[0][0])[i] = 0.f;
    for (int i = lane; i < 32 * 64; i += 32) (&sW1c[0][0])[i] = 0.f;
    for (int i = lane; i < 64 * 32; i += 32) (&sW1r[0][0])[i] = 0.f;
    for (int i = lane; i < 16 * 44; i += 32) (&sX[0][0])[i]   = 0.f;
    for (int i = lane; i < 64; i += 32) sb0[i] = (i < H0DIM) ? b0[i] : 0.f;
    sb1[lane] = (lane < H1DIM) ? b1[lane] : 0.f;
    swo[lane] = (lane < H1DIM) ? wo[lane] : 0.f;
    __syncthreads();

    // ---- stage: fill weights (coalesced, unconditional) ----
    for (int i = lane; i < FDIM * H0DIM; i += 32) {
        const int f = i / H0DIM, h = i - f * H0DIM;
        const float v = w0[i];
        sW0c[h][f] = v;
        sW0r[f][h] = v;
    }
    for (int i = lane; i < H0DIM * H1DIM; i += 32) {
        const int p = i / H1DIM, q = i - p * H1DIM;
        const float v = w1[i];
        sW1c[q][p] = v;
        sW1r[p][q] = v;
    }
    // ---- stage: X tile ----
    for (int i = lane; i < 16 * FDIM; i += 32) {
        const int r = i / FDIM, c = i - r * FDIM;
        const int row = tile * 16 + r;
        const int b = row / NAT_T, a = row - b * NAT_T + toff;
        sX[r][c] = image[((size_t)b * NATOM + a) * FDIM + c];
    }
    __syncthreads();

    // ---- GEMM1: H0 = X @ W0 + b0  (K padded to 44, N padded to 64) ----
    for (int n0 = 0; n0 < 64; n0 += 16) {
        v8f acc = {};
        const int n = n0 + lid;
        #pragma unroll
        for (int k = 0; k < 44; k += 4) {
            const int ka = k + 2 * hf;
            v2f a = *(const v2f*)&sX[lid][ka];
            v2f b = *(const v2f*)&sW0c[n][ka];
            acc = __builtin_amdgcn_wmma_f32_16x16x4_f32(false, a, false, b,
                                                        (short)0, acc, false, false);
        }
        const float bias = sb0[n];
        #pragma unroll
        for (int r = 0; r < 8; ++r)
            sH0[r + 8 * hf][n] = acc[r] + bias;
    }
    __syncthreads();

    // ---- GEMM2: H1 = H0 @ W1 + b1  (K=64, N padded to 32) ----
    for (int n0 = 0; n0 < 32; n0 += 16) {
        v8f acc = {};
        const int n = n0 + lid;
        #pragma unroll
        for (int k = 0; k < 64; k += 4) {
            const int ka = k + 2 * hf;
            v2f a = *(const v2f*)&sH0[lid][ka];
            v2f b = *(const v2f*)&sW1c[n][ka];
            acc = __builtin_amdgcn_wmma_f32_16x16x4_f32(false, a, false, b,
                                                        (short)0, acc, false, false);
        }
        const float bias = sb1[n];
        #pragma unroll
        for (int r = 0; r < 8; ++r)
            sH1[r + 8 * hf][n] = acc[r] + bias;
    }
    __syncthreads();

    // ---- Ei head: ei = relu(H1) @ wo + bo ----
    if (lane < 16) {
        const int row = tile * 16 + lane;
        const int b = row / NAT_T, a = row - b * NAT_T + toff;
        float e = bo[0];
        for (int j = 0; j < H1DIM; ++j) {
            const float h = sH1[lane][j];
            e += (h > 0.f ? h : 0.f) * swo[j];
        }
        ei_out[(size_t)b * NATOM + a] = e;
    }
    // dH1 = wo * relu'(H1): col = lane (zero-padded cols via swo pad)
    #pragma unroll
    for (int r = 0; r < 16; ++r)
        sdH1[r][lane] = (sH1[r][lane] > 0.f) ? swo[lane] : 0.f;
    __syncthreads();

    // ---- GEMM3: dH0 = dH1 @ W1^T  (K=32, N padded to 64) ----
    for (int n0 = 0; n0 < 64; n0 += 16) {
        v8f acc = {};
        const int n = n0 + lid;
        #pragma unroll
        for (int k = 0; k < 32; k += 4) {
            const int ka = k + 2 * hf;
            v2f a = *(const v2f*)&sdH1[lid][ka];
            v2f b = *(const v2f*)&sW1r[n][ka];
            acc = __builtin_amdgcn_wmma_f32_16x16x4_f32(false, a, false, b,
                                                        (short)0, acc, false, false);
        }
        #pragma unroll
        for (int r = 0; r < 8; ++r)
            sdH0[r + 8 * hf][n] = acc[r];
    }
    __syncthreads();

    // ---- GEMM4: g = dH0 @ W0^T  (K=64, N padded to 48) ----
    for (int n0 = 0; n0 < 48; n0 += 16) {
        v8f acc = {};
        const int n = n0 + lid;
        #pragma unroll
        for (int k = 0; k < 64; k += 4) {
            const int ka = k + 2 * hf;
            v2f a = *(const v2f*)&sdH0[lid][ka];
            v2f b = *(const v2f*)&sW0r[n][ka];
            acc = __builtin_amdgcn_wmma_f32_16x16x4_f32(false, a, false, b,
                                                        (short)0, acc, false, false);
        }
        #pragma unroll
        for (int r = 0; r < 8; ++r)
            sG[r + 8 * hf][n] = acc[r];
    }
    __syncthreads();

    // ---- coalesced store of the g tile ----
    for (int i = lane; i < 16 * FDIM; i += 32) {
        const int r = i / FDIM, c = i - r * FDIM;
        const int row = tile * 16 + r;
        const int b = row / NAT_T, a = row - b * NAT_T + toff;
        g[((size_t)b * NATOM + a) * FDIM + c] = sG[r][c];
    }
}

// ---------------------------------------------------------------------------
// Kernel 2: Etot[b] = sum_a Ei[b,a]  (deterministic tree reduction)
// ---------------------------------------------------------------------------
__global__ __launch_bounds__(256)
void etot_kernel(const float* __restrict__ ei, float* __restrict__ etot)
{
    __shared__ float s[256];
    const int b = blockIdx.x, t = threadIdx.x;
    float v = 0.f;
    for (int a = t; a < NATOM; a += 256) v += ei[(size_t)b * NATOM + a];
    s[t] = v;
    __syncthreads();
    for (int o = 128; o > 0; o >>= 1) {
        if (t < o) s[t] += s[t + o];
        __syncthreads();
    }
    if (t == 0) etot[b] = s[0];
}

// ---------------------------------------------------------------------------
// Kernel 3: Force[b,a,:] = sum_k mask * sum_f g[b,idx,f] * dfeat[b,a,k,f,:]
// One block per (b,a); 8 waves stride k; lanes stride the contiguous 126-float
// dfeat chunk -> coalesced 128B/wave loads; dfeat (348 MB) streamed once.
// ---------------------------------------------------------------------------
__global__ __launch_bounds__(256)
void force_kernel(const float* __restrict__ dfeat, const int* __restrict__ neighbor,
                  const float* __restrict__ g, float* __restrict__ force)
{
    __shared__ float s[256 * 3];
    const int ba  = blockIdx.x;             // 0 .. B*N-1
    const int b   = ba / NATOM;
    const int tid = threadIdx.x;
    const int lane = tid & 31;
    const int wid  = tid >> 5;              // 0..7

    float acc0 = 0.f, acc1 = 0.f, acc2 = 0.f;
    const size_t dbase = (size_t)ba * KNEI * FDIM * 3;
    const int*   nb = neighbor + (size_t)ba * KNEI;
    const float* gb = g + (size_t)b * NATOM * FDIM;

    for (int k = wid; k < KNEI; k += 8) {
        const int t = nb[k] - 1;            // 1-based neighbors
        if (t == 0) continue;               // mask: skip nei_index == 0
        const int idx = (t < 0) ? t + NATOM : t;   // torch negative wrap
        const float* gr = gb + (size_t)idx * FDIM;
        const float* df = dfeat + dbase + (size_t)k * (FDIM * 3);
        #pragma unroll
        for (int e = lane; e < FDIM * 3; e += 32) {
            const int f = e / 3;
            const int d = e - f * 3;
            const float v = gr[f] * df[e];
            if (d == 0) acc0 += v;
            else if (d == 1) acc1 += v;
            else acc2 += v;
        }
    }
    s[tid * 3 + 0] = acc0;
    s[tid * 3 + 1] = acc1;
    s[tid * 3 + 2] = acc2;
    __syncthreads();
    for (int o = 128; o > 0; o >>= 1) {
        if (tid < o) {
            s[tid * 3 + 0] += s[(tid + o) * 3 + 0];
            s[tid * 3 + 1] += s[(tid + o) * 3 + 1];
            s[tid * 3 + 2] += s[(tid + o) * 3 + 2];
        }
        __syncthreads();
    }
    if (tid < 3) force[(size_t)ba * 3 + tid] = s[tid];
}

// ---------------------------------------------------------------------------
extern "C" void kernel_launch(void* const* d_in, const int* in_sizes, int n_in,
                              void* d_out, int out_size, void* d_ws, size_t ws_size,
                              hipStream_t stream)
{
    (void)in_sizes; (void)n_in; (void)out_size; (void)ws_size;

    const float* image = (const float*)d_in[0];
    const float* dfeat = (const float*)d_in[1];
    // params tuple flattened leaf-order: type0 (w0,b0,w1,b1,wo,bo), then type1
    const float* w0a = (const float*)d_in[2],  *b0a = (const float*)d_in[3];
    const float* w1a = (const float*)d_in[4],  *b1a = (const float*)d_in[5];
    const float* woa = (const float*)d_in[6],  *boa = (const float*)d_in[7];
    const float* w0b = (const float*)d_in[8],  *b0b = (const float*)d_in[9];
    const float* w1b = (const float*)d_in[10], *b1b = (const float*)d_in[11];
    const float* wob = (const float*)d_in[12], *bob = (const float*)d_in[13];
    const int* neighbor = (const int*)d_in[14];

    float* out   = (float*)d_out;
    float* force = out;                                  // [B,N,3]
    float* etot  = out + (size_t)BATCH * NATOM * 3;      // [B,1]
    float* ei    = etot + BATCH;                         // [B,N,1]
    float* g     = (float*)d_ws;                         // [B,N,F] scratch

    mlp_fwd_bwd<<<TILES_T, 32, 0, stream>>>(image, w0a, b0a, w1a, b1a, woa, boa,
                                            g, ei, 0);
    mlp_fwd_bwd<<<TILES_T, 32, 0, stream>>>(image, w0b, b0b, w1b, b1b, wob, bob,
                                            g, ei, NAT_T);
    etot_kernel<<<BATCH, 256, 0, stream>>>(ei, etot);
    force_kernel<<<BATCH * NATOM, 256, 0, stream>>>(dfeat, neighbor, g, force);
}